// Model2_78073915507055
// MI455X (gfx1250) — compile-verified
//
#include <hip/hip_runtime.h>

typedef __attribute__((ext_vector_type(2))) float v2f;
typedef __attribute__((ext_vector_type(8))) float v8f;

#define BDIM 128   // batch
#define NDIM 64    // N
#define PDIM 128   // P
#define TDIM 168   // T
#define NN   4096  // N*N

// ---------------------------------------------------------------------------
// Kernel 1: gv = g @ (weights^2)   (4096x4096 matvec, bandwidth bound)
//           F[r,t] = gv[r] * alphas[t]   (rank-1 outer product, written out)
// One wave (32 lanes) per row; float4 coalesced loads; butterfly reduce.
// ---------------------------------------------------------------------------
__global__ void k_gv_F(const float* __restrict__ g,
                       const float* __restrict__ w,
                       const float* __restrict__ alphas,
                       float* __restrict__ gv,
                       float* __restrict__ Fout) {
    const int lane = threadIdx.x & 31;
    const int wave = threadIdx.x >> 5;
    const int row  = blockIdx.x * 8 + wave;        // 512 blocks * 8 waves = 4096 rows
    const float* grow = g + (size_t)row * NN;

    float sum = 0.f;
    #pragma unroll 4
    for (int i = 0; i < NN / (32 * 4); ++i) {      // 32 float4 chunks per lane
        const int c4 = (lane + i * 32) * 4;
        const float4 gg = *(const float4*)(grow + c4);
        const float4 ww = *(const float4*)(w + c4);
        sum += gg.x * ww.x * ww.x;
        sum += gg.y * ww.y * ww.y;
        sum += gg.z * ww.z * ww.z;
        sum += gg.w * ww.w * ww.w;
    }
    // wave32 butterfly reduction -> every lane holds the row total
    #pragma unroll
    for (int off = 16; off > 0; off >>= 1)
        sum += __shfl_xor(sum, off, 32);

    if (lane == 0) gv[row] = sum;
    for (int t = lane; t < TDIM; t += 32)
        Fout[(size_t)row * TDIM + t] = sum * alphas[t];
}

// ---------------------------------------------------------------------------
// Kernel 2: y[b,j] = sum_p alphas[x_i[b,p]] * x[b,j,p]
// One block per b; x[b] staged in LDS with +1 padding (conflict-free).
// ---------------------------------------------------------------------------
__global__ void k_y(const float* __restrict__ x,
                    const int*   __restrict__ xi,
                    const float* __restrict__ alphas,
                    float* __restrict__ y) {
    __shared__ float a_s[PDIM];
    __shared__ float xs[NDIM * (PDIM + 1)];
    const int b   = blockIdx.x;
    const int tid = threadIdx.x;                   // 256 threads

    if (tid < PDIM) a_s[tid] = alphas[xi[b * PDIM + tid]];

    const float* xb = x + (size_t)b * NDIM * PDIM;
    for (int k = tid; k < NDIM * PDIM; k += 256) { // coalesced global read
        const int j = k >> 7;                      // /128
        const int p = k & 127;
        xs[j * (PDIM + 1) + p] = xb[k];
    }
    __syncthreads();

    if (tid < NDIM) {
        float sum = 0.f;
        #pragma unroll 8
        for (int p = 0; p < PDIM; ++p)
            sum += a_s[p] * xs[tid * (PDIM + 1) + p];
        y[b * NDIM + tid] = sum;
    }
}

// ---------------------------------------------------------------------------
// Kernel 3: Z = Y(128x64) x G^T(64x64) via V_WMMA_F32_16X16X4_F32 (full f32).
// One wave per 16x16 tile of Z; 16 K-steps of K=4.
//   A lane layout (16x4, MxK):  m = lane&15, k-pair = 2*(lane>>4)  -> v2f
//   B lane layout (4x16, KxN):  n = lane&15, k-pair = 2*(lane>>4)  -> v2f
//   C/D (16x16): element v of v8f holds row m = v + 8*(lane>>4), col n = lane&15
// B[k][n] = G^T[k][n] = G[n][k] = gv[n*64 + k].
// ---------------------------------------------------------------------------
__global__ void k_z(const float* __restrict__ y,
                    const float* __restrict__ gv,
                    float* __restrict__ Z) {
    const int lane = threadIdx.x;                  // 32 threads per block
    const int b0 = blockIdx.x * 16;                // 8 tiles over batch
    const int i0 = blockIdx.y * 16;                // 4 tiles over output dim
    const int mh = lane & 15;
    const int kh = (lane >> 4) * 2;

    v8f c = {};
    #pragma unroll
    for (int k0 = 0; k0 < NDIM; k0 += 4) {
        v2f a, bm;
        a.x  = y[(b0 + mh) * NDIM + k0 + kh];
        a.y  = y[(b0 + mh) * NDIM + k0 + kh + 1];
        bm.x = gv[(i0 + mh) * NDIM + k0 + kh];     // G[i0+n][k]
        bm.y = gv[(i0 + mh) * NDIM + k0 + kh + 1];
        c = __builtin_amdgcn_wmma_f32_16x16x4_f32(
                /*neg_a=*/false, a, /*neg_b=*/false, bm,
                /*c_mod=*/(short)0, c, /*reuse_a=*/false, /*reuse_b=*/false);
    }

    const int hi = lane >> 4;
    #pragma unroll
    for (int v = 0; v < 8; ++v)
        Z[(b0 + v + 8 * hi) * NDIM + (i0 + mh)] = c[v];
}

// ---------------------------------------------------------------------------
extern "C" void kernel_launch(void* const* d_in, const int* in_sizes, int n_in,
                              void* d_out, int out_size, void* d_ws, size_t ws_size,
                              hipStream_t stream) {
    (void)in_sizes; (void)n_in; (void)out_size; (void)ws_size;
    const float* x      = (const float*)d_in[0];   // [128,64,128] f32
    const int*   xi     = (const int*)  d_in[1];   // [128,128] i32
    const float* g      = (const float*)d_in[2];   // [4096,4096] f32
    const float* w      = (const float*)d_in[3];   // [4096,1] f32
    const float* alphas = (const float*)d_in[4];   // [1,168] f32

    float* Z  = (float*)d_out;                     // [128,64]
    float* F  = (float*)d_out + BDIM * NDIM;       // [4096,168]
    float* gv = (float*)d_ws;                      // [4096]
    float* yb = gv + NN;                           // [128,64]

    k_gv_F<<<NN / 8, 256, 0, stream>>>(g, w, alphas, gv, F);
    k_y  <<<BDIM,    256, 0, stream>>>(x, xi, alphas, yb);
    dim3 zgrid(BDIM / 16, NDIM / 16);
    k_z  <<<zgrid,    32, 0, stream>>>(yb, gv, Z);
}